// HGR_86406152061405
// MI455X (gfx1250) — compile-verified
//
#include <hip/hip_runtime.h>
#include <hip/hip_bf16.h>
#include <math.h>

#define NUSERS 100000
#define NITEMS 50000
#define NGROUPS 4096
#define EMBD 64
#define MAXLEN 16
#define NTOT (NUSERS + NITEMS)

typedef __attribute__((ext_vector_type(2))) float v2f;
typedef __attribute__((ext_vector_type(8))) float v8f;

// ---------------------------------------------------------------------------
// Elementwise helpers
// ---------------------------------------------------------------------------
__global__ void k_init_ui(const float* __restrict__ user_t, const float* __restrict__ item_t,
                          float* __restrict__ cur, float* __restrict__ acc,
                          int n_user_f, int total_f) {
    for (int i = blockIdx.x * blockDim.x + threadIdx.x; i < total_f; i += gridDim.x * blockDim.x) {
        float v = (i < n_user_f) ? user_t[i] : item_t[i - n_user_f];
        cur[i] = v;
        acc[i] = v;
    }
}

__global__ void k_zero(float* __restrict__ p, int n) {
    for (int i = blockIdx.x * blockDim.x + threadIdx.x; i < n; i += gridDim.x * blockDim.x)
        p[i] = 0.0f;
}

__global__ void k_add(float* __restrict__ acc, const float* __restrict__ x, int n) {
    for (int i = blockIdx.x * blockDim.x + threadIdx.x; i < n; i += gridDim.x * blockDim.x)
        acc[i] += x[i];
}

__global__ void k_copy2(const float* __restrict__ src, float* __restrict__ a,
                        float* __restrict__ b, int n) {
    for (int i = blockIdx.x * blockDim.x + threadIdx.x; i < n; i += gridDim.x * blockDim.x) {
        float v = src[i];
        a[i] = v;
        b[i] = v;
    }
}

// ---------------------------------------------------------------------------
// Hypergraph SpMM: one wave32 per nonzero edge.
// lane j handles 2 consecutive floats of the 64-wide embedding row:
//   next[row] += val * cur[col]
// Gather is a coalesced 256B read; scatter uses native f32 atomics.
// ---------------------------------------------------------------------------
__global__ __launch_bounds__(256)
void k_spmm_edge(const int* __restrict__ rows, const int* __restrict__ cols,
                 const float* __restrict__ vals, const float* __restrict__ cur,
                 float* __restrict__ next, int nnz) {
    int wave = (blockIdx.x * blockDim.x + threadIdx.x) >> 5;
    int lane = threadIdx.x & 31;
    if (wave >= nnz) return;
    int r = rows[wave];
    int c = cols[wave];
    float v = vals[wave];
    const float2 cv = *(const float2*)(cur + (size_t)c * EMBD + lane * 2);
    float* dst = next + (size_t)r * EMBD + lane * 2;
    unsafeAtomicAdd(dst + 0, v * cv.x);
    unsafeAtomicAdd(dst + 1, v * cv.y);
}

// ---------------------------------------------------------------------------
// FP32 GEMM via V_WMMA_F32_16X16X4_F32.  C[M,N] = A[M,K] @ B[K,N], row-major,
// lda==K, ldb==ldc==N.  Requires M%64==0, N%64==0, K%16==0 (true for all
// call sites: 4096x64x4096).  256 threads = 8 waves; each wave owns a 16x32
// C strip of the 64x64 block tile.  No divergence => EXEC all ones for WMMA.
// ---------------------------------------------------------------------------
#define BM 64
#define BN 64
#define BK 16

__global__ __launch_bounds__(256, 2)
void gemm_f32_wmma(const float* __restrict__ A, const float* __restrict__ B,
                   float* __restrict__ C, int M, int N, int K) {
    __shared__ float As[BM][BK + 1];   // stride 17 words -> conflict-free column reads
    __shared__ float Bs[BK][BN + 4];

    const int tid   = threadIdx.x;
    const int wave  = tid >> 5;        // 0..7
    const int lane  = tid & 31;
    const int wm    = wave >> 1;       // 0..3 : M sub-tile
    const int wn    = wave & 1;        // 0..1 : N sub-tile (32 wide)
    const int laneM = lane & 15;
    const int hi    = lane >> 4;       // 0 or 1 (half-wave)

    const int m0 = blockIdx.y * BM;
    const int n0 = blockIdx.x * BN;

    v8f c0 = {};
    v8f c1 = {};

    const int arow = tid >> 2;          // 0..63
    const int acol = (tid & 3) * 4;     // 0,4,8,12
    const int brow = tid >> 4;          // 0..15
    const int bcol = (tid & 15) * 4;    // 0..60

    for (int k0 = 0; k0 < K; k0 += BK) {
        // cooperative, coalesced tile loads (float4 per thread)
        const float4 av = *(const float4*)(A + (size_t)(m0 + arow) * K + k0 + acol);
        As[arow][acol + 0] = av.x;
        As[arow][acol + 1] = av.y;
        As[arow][acol + 2] = av.z;
        As[arow][acol + 3] = av.w;
        const float4 bv = *(const float4*)(B + (size_t)(k0 + brow) * N + n0 + bcol);
        Bs[brow][bcol + 0] = bv.x;
        Bs[brow][bcol + 1] = bv.y;
        Bs[brow][bcol + 2] = bv.z;
        Bs[brow][bcol + 3] = bv.w;
        __syncthreads();

#pragma unroll
        for (int kk = 0; kk < BK; kk += 4) {
            // A 16x4 fragment: VGPR0 = K{0|2}, VGPR1 = K{1|3} per half-wave
            v2f a;
            a.x = As[wm * 16 + laneM][kk + hi * 2 + 0];
            a.y = As[wm * 16 + laneM][kk + hi * 2 + 1];
            // B 4x16 fragments, rows striped across lanes
            v2f b0, b1;
            b0.x = Bs[kk + hi * 2 + 0][wn * 32 + laneM];
            b0.y = Bs[kk + hi * 2 + 1][wn * 32 + laneM];
            b1.x = Bs[kk + hi * 2 + 0][wn * 32 + 16 + laneM];
            b1.y = Bs[kk + hi * 2 + 1][wn * 32 + 16 + laneM];
            c0 = __builtin_amdgcn_wmma_f32_16x16x4_f32(false, a, false, b0, (short)0, c0, false, false);
            c1 = __builtin_amdgcn_wmma_f32_16x16x4_f32(false, a, false, b1, (short)0, c1, false, false);
        }
        __syncthreads();
    }

    // C layout: VGPR r -> rows {r, r+8} per half-wave, cols = laneM
#pragma unroll
    for (int r = 0; r < 8; ++r) {
        int row = m0 + wm * 16 + hi * 8 + r;
        int col = n0 + wn * 32 + laneM;
        C[(size_t)row * N + col]      = c0[r];
        C[(size_t)row * N + col + 16] = c1[r];
    }
}

// ---------------------------------------------------------------------------
// Fused head: attention MLP + masked softmax + weighted member sum + group
// add + pred MLP + sigmoid.  One block (64 threads) per batch row.
// ---------------------------------------------------------------------------
__global__ __launch_bounds__(64)
void k_fused_head(const int* __restrict__ group_inputs, const int* __restrict__ item_inputs,
                  const int* __restrict__ member_masked, const float* __restrict__ mask,
                  const float* __restrict__ ui_acc, const float* __restrict__ g_acc,
                  const float* __restrict__ w1, const float* __restrict__ b1,
                  const float* __restrict__ w2, const float* __restrict__ b2,
                  const float* __restrict__ pw1, const float* __restrict__ pb1,
                  const float* __restrict__ pw2, const float* __restrict__ pb2,
                  float* __restrict__ out) {
    const int b = blockIdx.x;
    const int t = threadIdx.x;  // 0..63

    __shared__ float mem[MAXLEN][EMBD];
    __shared__ float item[EMBD];
    __shared__ float part[MAXLEN][EMBD];
    __shared__ float scores[MAXLEN];
    __shared__ float wt[MAXLEN];
    __shared__ float newemb[3 * EMBD];
    __shared__ float hid[8];

    const int it = item_inputs[b];
    item[t] = ui_acc[(size_t)(NUSERS + it) * EMBD + t];
#pragma unroll
    for (int l = 0; l < MAXLEN; ++l) {
        int u = member_masked[b * MAXLEN + l];
        mem[l][t] = ui_acc[(size_t)u * EMBD + t];
    }
    __syncthreads();

    // h = relu([member, item] @ w1 + b1); part = h * w2
    for (int l = 0; l < MAXLEN; ++l) {
        float acc = b1[t];
        for (int j = 0; j < EMBD; ++j) acc += mem[l][j] * w1[j * EMBD + t];
        for (int j = 0; j < EMBD; ++j) acc += item[j] * w1[(EMBD + j) * EMBD + t];
        float h = fmaxf(acc, 0.0f);
        part[l][t] = h * w2[t];
    }
    __syncthreads();

    if (t < MAXLEN) {
        float s = b2[0];
        for (int j = 0; j < EMBD; ++j) s += part[t][j];
        if (mask[b * MAXLEN + t] > 0.0f) s = -INFINITY;
        scores[t] = s;
    }
    __syncthreads();

    if (t < MAXLEN) {
        float mx = -INFINITY;
        for (int l = 0; l < MAXLEN; ++l) mx = fmaxf(mx, scores[l]);
        float sum = 0.0f;
        for (int l = 0; l < MAXLEN; ++l) sum += expf(scores[l] - mx);
        wt[t] = expf(scores[t] - mx) / sum;
    }
    __syncthreads();

    // g_att + g_pure, build [element, group, item]
    float ga = 0.0f;
    for (int l = 0; l < MAXLEN; ++l) ga += wt[l] * mem[l][t];
    const int g = group_inputs[b];
    float ge = ga + g_acc[(size_t)g * EMBD + t];
    float ie = item[t];
    newemb[t]            = ge * ie;
    newemb[EMBD + t]     = ge;
    newemb[2 * EMBD + t] = ie;
    __syncthreads();

    if (t < 8) {
        float acc = pb1[t];
        for (int j = 0; j < 3 * EMBD; ++j) acc += newemb[j] * pw1[j * 8 + t];
        hid[t] = fmaxf(acc, 0.0f);
    }
    __syncthreads();

    if (t == 0) {
        float acc = pb2[0];
        for (int k = 0; k < 8; ++k) acc += hid[k] * pw2[k];
        out[b] = 1.0f / (1.0f + expf(-acc));
    }
}

// ---------------------------------------------------------------------------
// Host orchestration
// ---------------------------------------------------------------------------
extern "C" void kernel_launch(void* const* d_in, const int* in_sizes, int n_in,
                              void* d_out, int out_size, void* d_ws, size_t ws_size,
                              hipStream_t stream) {
    const int*   group_inputs  = (const int*)d_in[0];
    const int*   item_inputs   = (const int*)d_in[1];
    const int*   member_masked = (const int*)d_in[2];
    const float* mask          = (const float*)d_in[3];
    const int*   adj_rows      = (const int*)d_in[4];
    const int*   adj_cols      = (const int*)d_in[5];
    const float* adj_vals      = (const float*)d_in[6];
    const float* Dmat          = (const float*)d_in[7];
    const float* Amat          = (const float*)d_in[8];
    const float* user_table    = (const float*)d_in[9];
    const float* item_table    = (const float*)d_in[10];
    const float* group_table   = (const float*)d_in[11];
    const float* attn_w1       = (const float*)d_in[12];
    const float* attn_b1       = (const float*)d_in[13];
    const float* attn_w2       = (const float*)d_in[14];
    const float* attn_b2       = (const float*)d_in[15];
    const float* pred_w1       = (const float*)d_in[16];
    const float* pred_b1       = (const float*)d_in[17];
    const float* pred_w2       = (const float*)d_in[18];
    const float* pred_b2       = (const float*)d_in[19];
    float* out = (float*)d_out;

    const int nnz  = in_sizes[4];
    const int NF   = NTOT * EMBD;        // 9,600,000 floats
    const int GF   = NGROUPS * EMBD;     // 262,144 floats
    const int BSZ  = in_sizes[0];        // 1024

    // carve workspace
    char* ws = (char*)d_ws;
    size_t off = 0;
    auto carve = [&](size_t bytes) -> float* {
        float* p = (float*)(ws + off);
        off += (bytes + 255) & ~(size_t)255;
        return p;
    };
    float* acc_ui  = carve((size_t)NF * 4);
    float* cur_ui  = carve((size_t)NF * 4);
    float* next_ui = carve((size_t)NF * 4);
    float* g_acc   = carve((size_t)GF * 4);
    float* g_cur   = carve((size_t)GF * 4);
    float* g_next  = carve((size_t)GF * 4);
    float* g_tmp   = carve((size_t)GF * 4);

    const int T = 256;
    const int ew_blocks = 2048;

    // ---- hypergraph conv: acc = sum of 3 propagation layers + input ----
    k_init_ui<<<ew_blocks, T, 0, stream>>>(user_table, item_table, cur_ui, acc_ui,
                                           NUSERS * EMBD, NF);
    for (int layer = 0; layer < 3; ++layer) {
        k_zero<<<ew_blocks, T, 0, stream>>>(next_ui, NF);
        int spmm_blocks = (int)(((long long)nnz * 32 + T - 1) / T);
        k_spmm_edge<<<spmm_blocks, T, 0, stream>>>(adj_rows, adj_cols, adj_vals,
                                                   cur_ui, next_ui, nnz);
        k_add<<<ew_blocks, T, 0, stream>>>(acc_ui, next_ui, NF);
        float* t2 = cur_ui; cur_ui = next_ui; next_ui = t2;
    }

    // ---- group conv: cur = D @ (A @ cur) per layer (assoc. reorder of DA@cur) ----
    k_copy2<<<256, T, 0, stream>>>(group_table, g_acc, g_cur, GF);
    dim3 ggrid(EMBD / BN, NGROUPS / BM);   // (1, 64)
    for (int layer = 0; layer < 3; ++layer) {
        gemm_f32_wmma<<<ggrid, T, 0, stream>>>(Amat, g_cur, g_tmp, NGROUPS, EMBD, NGROUPS);
        gemm_f32_wmma<<<ggrid, T, 0, stream>>>(Dmat, g_tmp, g_next, NGROUPS, EMBD, NGROUPS);
        k_add<<<256, T, 0, stream>>>(g_acc, g_next, GF);
        float* t2 = g_cur; g_cur = g_next; g_next = t2;
    }

    // ---- fused attention + prediction head ----
    k_fused_head<<<BSZ, 64, 0, stream>>>(group_inputs, item_inputs, member_masked, mask,
                                         acc_ui, g_acc,
                                         attn_w1, attn_b1, attn_w2, attn_b2,
                                         pred_w1, pred_b1, pred_w2, pred_b2,
                                         out);
}